// GCN_48558900249069
// MI455X (gfx1250) — compile-verified
//
#include <hip/hip_runtime.h>

#define N_NODES 100000
#define N_EDGES 1200000
#define DIN     128
#define HDIM    64
#define OUTC    10
#define NGRAPH  64
#define EPSBN   1e-5f
#define NLAYERS 4

typedef __attribute__((ext_vector_type(2))) float v2f;
typedef __attribute__((ext_vector_type(8))) float v8f;

#define WMMA_F32(a, b, c) \
    __builtin_amdgcn_wmma_f32_16x16x4_f32(false, (a), false, (b), (short)0, (c), false, false)

// ---------------------------------------------------------------- degree prep
__global__ __launch_bounds__(256) void init_deg_kernel(float* deg) {
    int n = blockIdx.x * blockDim.x + threadIdx.x;
    if (n < N_NODES) deg[n] = 1.0f;   // self-loop
}

__global__ __launch_bounds__(256) void deg_count_kernel(const int* __restrict__ dst,
                                                        float* __restrict__ deg) {
    int e = blockIdx.x * blockDim.x + threadIdx.x;
    if (e < N_EDGES) unsafeAtomicAdd(&deg[dst[e]], 1.0f);
}

__global__ __launch_bounds__(256) void deg_finalize_kernel(const float* __restrict__ deg,
                                                           float* __restrict__ dinv_sqrt,
                                                           float* __restrict__ inv_deg) {
    int n = blockIdx.x * blockDim.x + threadIdx.x;
    if (n < N_NODES) {
        float d = deg[n];
        dinv_sqrt[n] = rsqrtf(d);
        inv_deg[n]   = 1.0f / d;
    }
}

__global__ __launch_bounds__(256) void edge_norm_kernel(const int* __restrict__ src,
                                                        const int* __restrict__ dst,
                                                        const float* __restrict__ dinv_sqrt,
                                                        float* __restrict__ norm) {
    int e = blockIdx.x * blockDim.x + threadIdx.x;
    if (e < N_EDGES) norm[e] = dinv_sqrt[src[e]] * dinv_sqrt[dst[e]];
}

// ---------------------------------------------------------------- WMMA GEMM
// out[N,64] = A[N,K] @ W[K,64], fp32 via v_wmma_f32_16x16x4_f32.
// Block = 256 threads = 8 waves; wave w owns a 32x64 output tile (2 M-tiles x 4 N-tiles,
// 8 v8f accumulators). W is staged in LDS packed as K-pairs so each B fragment is one
// contiguous 8-byte LDS load: sW[(k/2)*128 + n*2 + (k&1)] = W[k][n].
__global__ __launch_bounds__(256) void gemm_wmma_kernel(const float* __restrict__ A,
                                                        const float* __restrict__ W,
                                                        float* __restrict__ out,
                                                        int K) {
    __shared__ float sW[(DIN / 2) * 128];   // 64*128 floats = 32 KB max (K<=128)
    const int tid = threadIdx.x;
    for (int i = tid; i < K * HDIM; i += 256) {
        const int k = i >> 6, n = i & 63;
        sW[(k >> 1) * 128 + n * 2 + (k & 1)] = W[i];
    }
    __syncthreads();

    const int wave = tid >> 5;
    const int lane = tid & 31;
    const int row0 = blockIdx.x * 256 + wave * 32;
    if (row0 >= N_NODES) return;      // wave-uniform: EXEC stays all-ones

    const int lh = lane >> 4;         // 0: K-pair {k,k+1} | 1: K-pair {k+2,k+3}
    const int ll = lane & 15;         // M index (A) / N index (B,C)

    v8f c00 = {}, c01 = {}, c02 = {}, c03 = {};   // rows row0..row0+15
    v8f c10 = {}, c11 = {}, c12 = {}, c13 = {};   // rows row0+16..row0+31

    const float* arow0 = A + (size_t)(row0 + ll) * K;
    const float* arow1 = arow0 + (size_t)16 * K;

    for (int k = 0; k < K; k += 4) {
        // A fragments: 16x4, lane ll holds row M=ll, K = k + 2*lh + {0,1}
        float2 a0v = *(const float2*)(arow0 + k + 2 * lh);
        float2 a1v = *(const float2*)(arow1 + k + 2 * lh);
        v2f a0; a0.x = a0v.x; a0.y = a0v.y;
        v2f a1; a1.x = a1v.x; a1.y = a1v.y;

        // B fragments: 4x16 slices of W; packed layout -> one b64 LDS load each
        const float* wb = sW + ((k >> 1) + lh) * 128 + ll * 2;
        const v2f b0 = *(const v2f*)(wb);
        const v2f b1 = *(const v2f*)(wb + 32);
        const v2f b2 = *(const v2f*)(wb + 64);
        const v2f b3 = *(const v2f*)(wb + 96);

        c00 = WMMA_F32(a0, b0, c00);
        c01 = WMMA_F32(a0, b1, c01);
        c02 = WMMA_F32(a0, b2, c02);
        c03 = WMMA_F32(a0, b3, c03);
        c10 = WMMA_F32(a1, b0, c10);
        c11 = WMMA_F32(a1, b1, c11);
        c12 = WMMA_F32(a1, b2, c12);
        c13 = WMMA_F32(a1, b3, c13);
    }

    // C/D layout: VGPR r, lanes 0-15 -> M=r, lanes 16-31 -> M=r+8, column = ll
    float* o0 = out + (size_t)row0 * HDIM;
    float* o1 = o0 + (size_t)16 * HDIM;
    const int mbase = lh * 8;
#pragma unroll
    for (int r = 0; r < 8; ++r) {
        const size_t m = (size_t)(mbase + r) * HDIM;
        o0[m +  0 + ll] = c00[r];
        o0[m + 16 + ll] = c01[r];
        o0[m + 32 + ll] = c02[r];
        o0[m + 48 + ll] = c03[r];
        o1[m +  0 + ll] = c10[r];
        o1[m + 16 + ll] = c11[r];
        o1[m + 32 + ll] = c12[r];
        o1[m + 48 + ll] = c13[r];
    }
}

// ---------------------------------------------------------------- per-layer pointwise
// agg = hw * inv_deg[node] + b[feat]   (self-loop term doubles as the zero-init)
__global__ __launch_bounds__(256) void selfloop_init_kernel(const float* __restrict__ hw,
                                                            const float* __restrict__ inv_deg,
                                                            const float* __restrict__ bias,
                                                            float* __restrict__ agg) {
    int t = blockIdx.x * blockDim.x + threadIdx.x;
    if (t < N_NODES * HDIM) {
        int n = t >> 6, f = t & 63;
        agg[t] = hw[t] * inv_deg[n] + bias[f];
    }
}

// agg[dst] += hw[src] * norm, float4-vectorized, 16 lanes per edge, native fp32 atomics
__global__ __launch_bounds__(256) void scatter_add_kernel(const float* __restrict__ hw,
                                                          const int* __restrict__ src,
                                                          const int* __restrict__ dst,
                                                          const float* __restrict__ norm,
                                                          float* __restrict__ agg) {
    long long t = (long long)blockIdx.x * blockDim.x + threadIdx.x;
    if (t >= (long long)N_EDGES * 16) return;
    const int e  = (int)(t >> 4);
    const int f4 = ((int)t & 15) << 2;
    const float nrm = norm[e];
    const float4 v = *(const float4*)(hw + (size_t)src[e] * HDIM + f4);
    float* a = agg + (size_t)dst[e] * HDIM + f4;
    unsafeAtomicAdd(a + 0, v.x * nrm);
    unsafeAtomicAdd(a + 1, v.y * nrm);
    unsafeAtomicAdd(a + 2, v.z * nrm);
    unsafeAtomicAdd(a + 3, v.w * nrm);
}

// h = relu(gamma*(agg-mean)*rsqrt(var+eps)+beta)
__global__ __launch_bounds__(256) void bn_relu_kernel(const float* __restrict__ agg,
                                                      const float* __restrict__ gamma,
                                                      const float* __restrict__ beta,
                                                      const float* __restrict__ mean,
                                                      const float* __restrict__ var,
                                                      float* __restrict__ hout) {
    int t = blockIdx.x * blockDim.x + threadIdx.x;
    if (t < N_NODES * HDIM) {
        int f = t & 63;
        float s = gamma[f] * rsqrtf(var[f] + EPSBN);
        float v = (agg[t] - mean[f]) * s + beta[f];
        hout[t] = fmaxf(v, 0.0f);
    }
}

// ---------------------------------------------------------------- pooling + classifier
__global__ __launch_bounds__(256) void pool_init_kernel(float* pooled) {
    int t = blockIdx.x * blockDim.x + threadIdx.x;
    if (t < NGRAPH * HDIM) pooled[t] = 0.0f;   // valid: features are post-ReLU (>= 0)
}

// segment_max via int atomicMax (bit pattern of non-negative floats is order-preserving)
__global__ __launch_bounds__(256) void pool_max_kernel(const float* __restrict__ h,
                                                       const int* __restrict__ batch,
                                                       int* __restrict__ pooled_i) {
    int t = blockIdx.x * blockDim.x + threadIdx.x;
    if (t < N_NODES * HDIM) {
        int n = t >> 6, f = t & 63;
        atomicMax(&pooled_i[batch[n] * HDIM + f], __float_as_int(h[t]));
    }
}

__global__ __launch_bounds__(256) void classifier_kernel(const float* __restrict__ pooled,
                                                         const float* __restrict__ Wc,
                                                         const float* __restrict__ bc,
                                                         float* __restrict__ out) {
    int t = blockIdx.x * blockDim.x + threadIdx.x;
    if (t >= NGRAPH * OUTC) return;
    int g = t / OUTC, o = t % OUTC;
    float acc = bc[o];
#pragma unroll
    for (int f = 0; f < HDIM; ++f) acc += pooled[g * HDIM + f] * Wc[f * OUTC + o];
    out[t] = acc;
}

// ---------------------------------------------------------------- launch
extern "C" void kernel_launch(void* const* d_in, const int* in_sizes, int n_in,
                              void* d_out, int out_size, void* d_ws, size_t ws_size,
                              hipStream_t stream) {
    const float* x     = (const float*)d_in[0];
    const int*   ei    = (const int*)d_in[1];
    const int*   src   = ei;
    const int*   dst   = ei + N_EDGES;
    const int*   batch = (const int*)d_in[2];

    const float *W[NLAYERS], *b[NLAYERS], *gm[NLAYERS], *bt[NLAYERS], *mn[NLAYERS], *vr[NLAYERS];
    for (int i = 0; i < NLAYERS; ++i) {
        int base = 3 + 6 * i;
        W[i]  = (const float*)d_in[base + 0];
        b[i]  = (const float*)d_in[base + 1];
        gm[i] = (const float*)d_in[base + 2];
        bt[i] = (const float*)d_in[base + 3];
        mn[i] = (const float*)d_in[base + 4];
        vr[i] = (const float*)d_in[base + 5];
    }
    const float* Wc = (const float*)d_in[3 + 6 * NLAYERS];
    const float* bc = (const float*)d_in[4 + 6 * NLAYERS];

    // workspace carve-up (floats)
    float* ws        = (float*)d_ws;
    float* deg       = ws;                                   // N
    float* dinv_sqrt = deg + N_NODES;                        // N
    float* inv_deg   = dinv_sqrt + N_NODES;                  // N
    float* norm      = inv_deg + N_NODES;                    // E
    float* hbuf      = norm + N_EDGES;                       // N*H
    float* hw        = hbuf + (size_t)N_NODES * HDIM;        // N*H
    float* agg       = hw   + (size_t)N_NODES * HDIM;        // N*H
    float* pooled    = agg  + (size_t)N_NODES * HDIM;        // G*H

    const int TB = 256;
    const int nBlkN  = (N_NODES + TB - 1) / TB;
    const int nBlkE  = (N_EDGES + TB - 1) / TB;
    const int nBlkNH = (N_NODES * HDIM + TB - 1) / TB;
    const int nBlkSc = (int)(((long long)N_EDGES * 16 + TB - 1) / TB);
    const int nBlkGemm = (N_NODES + 255) / 256;   // 256 rows per block (8 waves x 32 rows)

    // degree / norm (computed once, reused by all layers)
    init_deg_kernel<<<nBlkN, TB, 0, stream>>>(deg);
    deg_count_kernel<<<nBlkE, TB, 0, stream>>>(dst, deg);
    deg_finalize_kernel<<<nBlkN, TB, 0, stream>>>(deg, dinv_sqrt, inv_deg);
    edge_norm_kernel<<<nBlkE, TB, 0, stream>>>(src, dst, dinv_sqrt, norm);

    const float* hin = x;
    for (int i = 0; i < NLAYERS; ++i) {
        const int K = (i == 0) ? DIN : HDIM;
        gemm_wmma_kernel<<<nBlkGemm, TB, 0, stream>>>(hin, W[i], hw, K);
        selfloop_init_kernel<<<nBlkNH, TB, 0, stream>>>(hw, inv_deg, b[i], agg);
        scatter_add_kernel<<<nBlkSc, TB, 0, stream>>>(hw, src, dst, norm, agg);
        bn_relu_kernel<<<nBlkNH, TB, 0, stream>>>(agg, gm[i], bt[i], mn[i], vr[i], hbuf);
        hin = hbuf;
    }

    pool_init_kernel<<<(NGRAPH * HDIM + TB - 1) / TB, TB, 0, stream>>>(pooled);
    pool_max_kernel<<<nBlkNH, TB, 0, stream>>>(hbuf, batch, (int*)pooled);
    classifier_kernel<<<(NGRAPH * OUTC + TB - 1) / TB, TB, 0, stream>>>(pooled, Wc, bc, (float*)d_out);
}